// ValueCrossAttn_84430467105063
// MI455X (gfx1250) — compile-verified
//
#include <hip/hip_runtime.h>

typedef __attribute__((ext_vector_type(2))) float v2f;
typedef __attribute__((ext_vector_type(8))) float v8f;

#define QUERY_DIM 256
#define CONTEXT_DIM 16
#define INNER 512
#define BATCH 2
// output: [2, 256, 256, 256] f32; each (b,c) slab = 65536 contiguous floats

// ---------------------------------------------------------------------------
// Kernel 1: y[b][c] = sum_i Wo[c][i] * (sum_j Wv[i][j] * context[b][j]) + bo[c]
// One block, 512 threads = 16 wave32. Stage B = V_WMMA_F32_16X16X4_F32.
// ---------------------------------------------------------------------------
__global__ __launch_bounds__(512, 1)
void vca_head_kernel(const float* __restrict__ context,  // [2][16]
                     const float* __restrict__ Wv,       // [512][16]
                     const float* __restrict__ Wo,       // [256][512]
                     const float* __restrict__ bo,       // [256]
                     float* __restrict__ y) {            // [2][256] (workspace)
  // v padded to 16 "batch" rows (rows 2..15 = 0) so the WMMA B operand needs
  // no per-lane predication. Row stride 514 floats: even (8B-aligned b64 LDS
  // loads) and 514 % 64 == 2 so the 16 rows map to distinct bank pairs.
  __shared__ float vsh[16][514];

  const int t = threadIdx.x;

  // ---- Stage A: v[b][i] = context[b][:] . Wv[i][:]  (thread t handles i=t)
  float c0 = 0.f, c1 = 0.f;
#pragma unroll
  for (int j = 0; j < CONTEXT_DIM; ++j) {
    const float w = Wv[t * CONTEXT_DIM + j];
    c0 += context[j] * w;
    c1 += context[CONTEXT_DIM + j] * w;
  }
  vsh[0][t] = c0;
  vsh[1][t] = c1;
#pragma unroll
  for (int n = 2; n < 16; ++n) vsh[n][t] = 0.f;
  __syncthreads();

  // ---- Stage B: y[0:2][c] = Wo[c][:] . v[0:2][:]  via wmma f32 16x16x4
  // Wave w owns M-tile rows m0..m0+15 (c-dimension); N columns = batch (0,1
  // valid, 2..15 are zero padding).
  const int wave = t >> 5;
  const int lane = t & 31;
  const int h  = lane >> 4;   // half-wave: selects K pair {2h, 2h+1}
  const int lL = lane & 15;   // A: row M; B: column N
  const int m0 = wave * 16;

  const float* wo_row = Wo + (size_t)(m0 + lL) * INNER;
  const float* v_row  = &vsh[lL][0];

  v8f acc = {};
#pragma unroll 4
  for (int kk = 0; kk < INNER; kk += 4) {
    const int k = kk + 2 * h;
    // A 16x4 f32 layout: VGPR0 = K {0 | 2}, VGPR1 = K {1 | 3} per half-wave.
    v2f a = *(const v2f*)(wo_row + k);
    // B 4x16 f32 layout: VGPR0 = row K {0 | 2}, VGPR1 = row K {1 | 3};
    // B[k][n] = v[n][k] with n = lane&15.
    v2f b = *(const v2f*)(v_row + k);
    acc = __builtin_amdgcn_wmma_f32_16x16x4_f32(
        /*neg_a=*/false, a, /*neg_b=*/false, b,
        /*c_mod=*/(short)0, acc, /*reuse_a=*/false, /*reuse_b=*/false);
  }

  // D layout: acc[r] @ lane(h,lL) = D[m0 + r + 8h][N = lL]. Keep N = batch 0,1.
  if (lL < BATCH) {
#pragma unroll
    for (int r = 0; r < 8; ++r) {
      const int c = m0 + r + 8 * h;
      y[lL * QUERY_DIM + c] = acc[r] + bo[c];
    }
  }
}

// ---------------------------------------------------------------------------
// Kernel 2: broadcast y[b][c] over the 65536-pixel slab. Pure b128 stores.
// 8192 blocks x 256 threads; block writes 4096 consecutive floats; 16 blocks
// per slab -> slab id is block-uniform (scalar y load), stores fully coalesced.
// Output (134 MB) fits in the 192 MB L2, so default RT stores are optimal.
// ---------------------------------------------------------------------------
__global__ __launch_bounds__(256, 8)
void vca_broadcast_kernel(const float* __restrict__ y, float4* __restrict__ out4) {
  const int slab = blockIdx.x >> 4;          // (b*256 + c), uniform per block
  const float v = y[slab];
  const float4 val = make_float4(v, v, v, v);
  float4* base = out4 + (size_t)blockIdx.x * 1024;
#pragma unroll
  for (int k = 0; k < 4; ++k)
    base[(size_t)k * 256 + threadIdx.x] = val;
}

extern "C" void kernel_launch(void* const* d_in, const int* in_sizes, int n_in,
                              void* d_out, int out_size, void* d_ws, size_t ws_size,
                              hipStream_t stream) {
  // setup_inputs order: x=0, context=1, Wq=2, Wk=3, Wv=4, Wo=5, bo=6
  const float* context = (const float*)d_in[1];
  const float* Wv      = (const float*)d_in[4];
  const float* Wo      = (const float*)d_in[5];
  const float* bo      = (const float*)d_in[6];
  float* y = (float*)d_ws;  // 512 floats of scratch

  vca_head_kernel<<<1, 512, 0, stream>>>(context, Wv, Wo, bo, y);
  // 2*256*256*256 floats = 8,388,608 float4s = 8192 blocks * 256 thr * 4
  vca_broadcast_kernel<<<8192, 256, 0, stream>>>(y, (float4*)d_out);
}